// KernelSetConv_21689584845342
// MI455X (gfx1250) — compile-verified
//
#include <hip/hip_runtime.h>
#include <math.h>

typedef __attribute__((ext_vector_type(2))) float v2f;
typedef __attribute__((ext_vector_type(8))) float v8f;

#define K_OUT 16
#define F_DIM 32
#define E_DIM 8

__device__ __forceinline__ float wsum(float v) {
  v += __shfl_xor(v, 16, 32);
  v += __shfl_xor(v, 8, 32);
  v += __shfl_xor(v, 4, 32);
  v += __shfl_xor(v, 2, 32);
  v += __shfl_xor(v, 1, 32);
  return v;
}

__device__ __forceinline__ float rnorm_of(float ss) {
  float n = sqrtf(ss);
  return n > 1e-12f ? 1.0f / n : 0.0f;
}

template <int D> struct GeomT {
  static constexpr int OFF_NEI = F_DIM;                 // 32
  static constexpr int OFF_P   = F_DIM + F_DIM * D;     // 32 + 32d
  static constexpr int OFF_E   = OFF_P + 3 * D;         // + 3d
  static constexpr int L       = OFF_E + E_DIM * D;     // 32 + 43d
  static constexpr int LPAD    = (L + 3) & ~3;          // multiple of 4 for K=4 WMMA
};

// ---------------------------------------------------------------------------
// B preparation: Bt[l][k] (LPAD x 16), each segment of row k L2-normalized,
// neighbor/pos/edge segments pre-scaled by 1/D (the mean over d). One wave per k.
// ---------------------------------------------------------------------------
template <int D>
__global__ void prep_b_kernel(const float* __restrict__ Wc,
                              const float* __restrict__ Wn,
                              const float* __restrict__ Wp,
                              const float* __restrict__ We,
                              float* __restrict__ Bt) {
  using G = GeomT<D>;
  const int lane = threadIdx.x & 31;
  const int k = threadIdx.x >> 5;
  if (k >= K_OUT) return;
  const float invd = 1.0f / (float)D;
  {
    float v = Wc[k * F_DIM + lane];
    float rs = rnorm_of(wsum(v * v));
    Bt[lane * K_OUT + k] = v * rs;
  }
  for (int j = 0; j < D; ++j) {
    {
      float v = Wn[(k * D + j) * F_DIM + lane];
      float rs = rnorm_of(wsum(v * v)) * invd;
      Bt[(G::OFF_NEI + F_DIM * j + lane) * K_OUT + k] = v * rs;
    }
    {
      float v = (lane < 3) ? Wp[(k * D + j) * 3 + lane] : 0.0f;
      float rs = rnorm_of(wsum(v * v)) * invd;
      if (lane < 3) Bt[(G::OFF_P + 3 * j + lane) * K_OUT + k] = v * rs;
    }
    {
      float v = (lane < E_DIM) ? We[(k * D + j) * E_DIM + lane] : 0.0f;
      float rs = rnorm_of(wsum(v * v)) * invd;
      if (lane < E_DIM) Bt[(G::OFF_E + E_DIM * j + lane) * K_OUT + k] = v * rs;
    }
  }
  for (int l = G::L + lane; l < G::LPAD; l += 32)
    Bt[l * K_OUT + k] = 0.0f;
}

// ---------------------------------------------------------------------------
// Fused gather + normalize (into wave-private LDS) + f32 WMMA GEMM + full-row
// output write. One wave handles a 16-node x 16-template tile.
// ---------------------------------------------------------------------------
template <int D, int GRP>
__global__ void __launch_bounds__(128)
score_kernel(const float* __restrict__ x, const int* __restrict__ sel,
             const int* __restrict__ nei, const float* __restrict__ neip,
             const float* __restrict__ neie, const float* __restrict__ Bt,
             float* __restrict__ out, int nd) {
  using G = GeomT<D>;
  constexpr int WPB = 4;           // waves per block
  constexpr int STRIDE = G::LPAD;  // LDS row stride (floats)
  __shared__ float smem[WPB * 16 * STRIDE];

  const int lane = threadIdx.x & 31;
  const int wave = threadIdx.x >> 5;
  const int tile = blockIdx.x * WPB + wave;
  const int tiles = (nd + 15) >> 4;
  if (tile >= tiles) return;

  float* A = &smem[wave * 16 * STRIDE];

  // Phase 1: gather + L2-normalize 16 node rows into LDS (wave-private region).
  for (int r = 0; r < 16; ++r) {
    float* row = A + r * STRIDE;
    const int idx = tile * 16 + r;
    if (idx < nd) {
      const int si = sel[idx];
      {
        float v = x[(size_t)si * F_DIM + lane];
        row[lane] = v * rnorm_of(wsum(v * v));
      }
      for (int j = 0; j < D; ++j) {
        const int ni = nei[idx * D + j];
        {
          float v = x[(size_t)ni * F_DIM + lane];
          row[G::OFF_NEI + F_DIM * j + lane] = v * rnorm_of(wsum(v * v));
        }
        {
          float v = (lane < 3) ? neip[((size_t)idx * D + j) * 3 + lane] : 0.0f;
          float rs = rnorm_of(wsum(v * v));
          if (lane < 3) row[G::OFF_P + 3 * j + lane] = v * rs;
        }
        {
          float v = (lane < E_DIM) ? neie[((size_t)idx * D + j) * E_DIM + lane] : 0.0f;
          float rs = rnorm_of(wsum(v * v));
          if (lane < E_DIM) row[G::OFF_E + E_DIM * j + lane] = v * rs;
        }
      }
      for (int l = G::L + lane; l < G::LPAD; l += 32) row[l] = 0.0f;
    } else {
      for (int l = lane; l < G::LPAD; l += 32) row[l] = 0.0f;  // zero padding rows
    }
  }
  // Same-wave DS store->load dependencies are handled by compiler-inserted waits.

  // Phase 2: D = A(16xLPAD) * B(LPADx16) via chained V_WMMA_F32_16X16X4_F32.
  // A fragment: lane holds row (lane&15), K-slots {off, off+1}, off = (lane>>4)*2.
  // B fragment: lane holds col (lane&15), rows {kc+off, kc+off+1}.
  v8f acc = {};
  const int col = lane & 15;
  const int off = (lane >> 4) << 1;
  const float* arow = A + col * STRIDE;
  for (int kc = 0; kc < G::LPAD; kc += 4) {
    v2f a, b;
    a.x = arow[kc + off];
    a.y = arow[kc + off + 1];
    b.x = Bt[(kc + off) * K_OUT + col];
    b.y = Bt[(kc + off + 1) * K_OUT + col];
    acc = __builtin_amdgcn_wmma_f32_16x16x4_f32(false, a, false, b,
                                                (short)0, acc, false, false);
  }

  // Phase 3: each node appears in exactly one group's sel, so write the whole
  // 64-wide output row (scores in our 16-col block, zeros elsewhere).
  const int halfsel = lane >> 4;
#pragma unroll
  for (int i = 0; i < 8; ++i) {
    const int r = i + halfsel * 8;  // C/D layout: VGPR i, lanes>=16 => M = i+8
    const int idx = tile * 16 + r;
    if (idx < nd) {
      const int orow = sel[idx];
      float* po = out + (size_t)orow * 64 + col;
      const float sv = acc[i] * 0.25f;
#pragma unroll
      for (int g = 0; g < 4; ++g) po[g * K_OUT] = (g == GRP) ? sv : 0.0f;
    }
  }
}

// ---------------------------------------------------------------------------
template <int D, int GRP>
static void launch_group(void* const* d_in, const int* in_sizes, void* d_out,
                         void* d_ws, hipStream_t stream) {
  const float* x = (const float*)d_in[0];
  const int base = 1 + GRP * 8;
  const int* sel   = (const int*)d_in[base + 0];
  const int* nei   = (const int*)d_in[base + 1];
  const float* nep = (const float*)d_in[base + 2];
  const float* nee = (const float*)d_in[base + 3];
  const float* Wc  = (const float*)d_in[base + 4];
  const float* Wn  = (const float*)d_in[base + 5];
  const float* Wp  = (const float*)d_in[base + 6];
  const float* We  = (const float*)d_in[base + 7];
  float* Bt = (float*)((char*)d_ws + (size_t)GRP * 16384);  // 16KB per group
  float* out = (float*)d_out;
  const int nd = in_sizes[base + 0];

  prep_b_kernel<D><<<1, 512, 0, stream>>>(Wc, Wn, Wp, We, Bt);

  const int tiles = (nd + 15) / 16;
  const int blocks = (tiles + 3) / 4;  // 4 waves per block
  score_kernel<D, GRP><<<blocks, 128, 0, stream>>>(x, sel, nei, nep, nee, Bt, out, nd);
}

extern "C" void kernel_launch(void* const* d_in, const int* in_sizes, int n_in,
                              void* d_out, int out_size, void* d_ws, size_t ws_size,
                              hipStream_t stream) {
  (void)n_in; (void)out_size; (void)ws_size;
  launch_group<1, 0>(d_in, in_sizes, d_out, d_ws, stream);
  launch_group<2, 1>(d_in, in_sizes, d_out, d_ws, stream);
  launch_group<3, 2>(d_in, in_sizes, d_out, d_ws, stream);
  launch_group<4, 3>(d_in, in_sizes, d_out, d_ws, stream);
}